// GEARS_Model_68393059221715
// MI455X (gfx1250) — compile-verified
//
#include <hip/hip_runtime.h>

#define NG   5000
#define HID  64
#define GG   64
#define NE   100000
#define GN   (GG*NG)      // 320000
#define EPSB 1e-5f

typedef __attribute__((ext_vector_type(2))) float v2f;
typedef __attribute__((ext_vector_type(8))) float v8f;

// ---------------- small utility kernels ----------------

__global__ void k_renorm(const float* __restrict__ src, float* __restrict__ dst, int rows) {
  int r = blockIdx.x * blockDim.x + threadIdx.x;
  if (r >= rows) return;
  const float* s = src + r * HID;
  float acc = 0.f;
  for (int k = 0; k < HID; k++) { float v = s[k]; acc += v * v; }
  float n = sqrtf(acc);
  float sc = (n > 1.0f) ? (1.0f / (n + 1e-7f)) : 1.0f;
  float* d = dst + r * HID;
  for (int k = 0; k < HID; k++) d[k] = s[k] * sc;
}

__global__ void k_zero(float* p, int n) {
  int i = blockIdx.x * blockDim.x + threadIdx.x;
  if (i < n) p[i] = 0.f;
}

__global__ void k_scatter_deg(const int* __restrict__ ei, const float* __restrict__ ew,
                              float* deg, int ne) {
  int e = blockIdx.x * blockDim.x + threadIdx.x;
  if (e < ne) atomicAdd(&deg[ei[ne + e]], ew[e]);
}

__global__ void k_dinv(const float* deg, float* dinv, int n) {
  int i = blockIdx.x * blockDim.x + threadIdx.x;
  if (i < n) dinv[i] = rsqrtf(deg[i] + 1.0f);
}

__global__ void k_selfagg(const float* __restrict__ xn, const float* __restrict__ dinv,
                          float* agg, int rows) {
  int i = blockIdx.x * blockDim.x + threadIdx.x;
  if (i < rows * HID) { int r = i >> 6; agg[i] = dinv[r] * dinv[r] * xn[i]; }
}

__global__ void k_scatter_edges(const int* __restrict__ ei, const float* __restrict__ ew,
                                const float* __restrict__ dinv, const float* __restrict__ xn,
                                float* agg, int ne) {
  int e = blockIdx.x * blockDim.x + threadIdx.x;
  if (e >= ne) return;
  int r = ei[e], c = ei[ne + e];
  float coef = dinv[r] * ew[e] * dinv[c];
  const float* xs = xn + r * HID;
  float* ad = agg + c * HID;
  for (int h = 0; h < HID; h++) atomicAdd(&ad[h], coef * xs[h]);
}

// dst[r][j] = (add ? add[r][j] + alpha*lin : lin), lin = src[r]·w[j] + b[j]
__global__ void k_lin64(const float* __restrict__ src, const float* __restrict__ w,
                        const float* __restrict__ b, const float* __restrict__ add,
                        float alpha, float* __restrict__ dst, int rows) {
  int i = blockIdx.x * blockDim.x + threadIdx.x;
  if (i >= rows * HID) return;
  int r = i >> 6, j = i & 63;
  const float* s = src + r * HID;
  const float* wj = w + j * HID;
  float acc = b[j];
  for (int k = 0; k < HID; k++) acc += s[k] * wj[k];
  dst[i] = add ? (add[i] + alpha * acc) : acc;
}

// one block per column; deterministic tree reduction over `rows`
__global__ void k_bnstats(const float* __restrict__ src, int rows,
                          const float* __restrict__ g, const float* __restrict__ b,
                          float* s_out, float* t_out) {
  __shared__ float ss[256], sq[256];
  int col = blockIdx.x, tid = threadIdx.x;
  float S = 0.f, Q = 0.f;
  for (int r = tid; r < rows; r += 256) { float v = src[r * HID + col]; S += v; Q += v * v; }
  ss[tid] = S; sq[tid] = Q; __syncthreads();
  for (int st = 128; st > 0; st >>= 1) {
    if (tid < st) { ss[tid] += ss[tid + st]; sq[tid] += sq[tid + st]; }
    __syncthreads();
  }
  if (tid == 0) {
    float m = ss[0] / rows, var = sq[0] / rows - m * m;
    float s = g[col] * rsqrtf(var + EPSB);
    s_out[col] = s; t_out[col] = b[col] - m * s;
  }
}

__global__ void k_bnapply(const float* __restrict__ src, float* __restrict__ dst, int n,
                          const float* __restrict__ s, const float* __restrict__ t, int relu) {
  int i = blockIdx.x * blockDim.x + threadIdx.x;
  if (i >= n) return;
  int c = i & 63;
  float v = src[i] * s[c] + t[c];
  dst[i] = relu ? fmaxf(v, 0.f) : v;
}

__global__ void k_pertsum(const float* __restrict__ x, const float* __restrict__ pg,
                          float* __restrict__ ps) {
  int g = blockIdx.x, h = threadIdx.x;
  float acc = 0.f;
  for (int n = 0; n < NG; n++) {
    float f = x[(size_t)(g * NG + n) * 2 + 1];
    if (f > 0.5f) acc += pg[n * HID + h];
  }
  ps[g * HID + h] = acc;
}

// on-the-fly column stats over the [GN,64] "combined" batch (modeY applies 1st BN+relu)
__global__ void k_bigstats(const float* __restrict__ baseF, const float* __restrict__ embT,
                           const float* __restrict__ x, const float* __restrict__ pww,
                           const float* __restrict__ pwb, const float* __restrict__ s1,
                           const float* __restrict__ t1, int modeY, float* __restrict__ partial) {
  __shared__ float sh[256], shq[256];
  int tid = threadIdx.x;
  int col = tid & 63, rg = tid >> 6;
  float S = 0.f, Q = 0.f;
  for (int row = blockIdx.x * 4 + rg; row < GN; row += gridDim.x * 4) {
    int g = row / NG, n = row - g * NG;
    float f = x[(size_t)2 * row + 1];
    float v = baseF[n * HID + col] + embT[g * HID + col] + pww[col] * f + pwb[col];
    if (modeY) v = fmaxf(v * s1[col] + t1[col], 0.f);
    S += v; Q += v * v;
  }
  sh[tid] = S; shq[tid] = Q; __syncthreads();
  if (tid < 64) {
    float s = sh[tid] + sh[tid + 64] + sh[tid + 128] + sh[tid + 192];
    float q = shq[tid] + shq[tid + 64] + shq[tid + 128] + shq[tid + 192];
    partial[blockIdx.x * 128 + tid] = s;
    partial[blockIdx.x * 128 + 64 + tid] = q;
  }
}

// fixed-order merge of per-block partials -> (scale, shift) for a BN layer
__global__ void k_bnfinal(const float* __restrict__ partial, int nblocks, int ncols, float cnt,
                          const float* __restrict__ g, const float* __restrict__ b,
                          float* s_out, float* t_out) {
  int col = threadIdx.x;
  if (col >= ncols) return;
  float S = 0.f, Q = 0.f;
  for (int k = 0; k < nblocks; k++) {
    S += partial[k * 2 * ncols + col];
    Q += partial[k * 2 * ncols + ncols + col];
  }
  float m = S / cnt, var = Q / cnt - m * m;
  float s = g[col] * rsqrtf(var + EPSB);
  s_out[col] = s; t_out[col] = b[col] - m * s;
}

// ---------------- fused recovery MLP (WMMA f32 16x16x4) ----------------
// MODE 0: act -> l1 pre-BN column stats (128 cols)
// MODE 1: act -> l1 -> bn1 -> relu -> l2 pre-BN column stats (64 cols)
// MODE 2: full chain -> out -> w[g,n] = out·indv_w1[n] + indv_b1[n]
template <int MODE>
__global__ void __launch_bounds__(128) k_recovery(
    const float* __restrict__ baseF, const float* __restrict__ embT, const float* __restrict__ x,
    const float* __restrict__ pww, const float* __restrict__ pwb,
    const float* __restrict__ s1, const float* __restrict__ t1,
    const float* __restrict__ s2, const float* __restrict__ t2,
    const float* __restrict__ w1, const float* __restrict__ b1,
    const float* __restrict__ r1s, const float* __restrict__ r1t,
    const float* __restrict__ w2, const float* __restrict__ b2,
    const float* __restrict__ r2s, const float* __restrict__ r2t,
    const float* __restrict__ iw1, const float* __restrict__ ib1,
    float* __restrict__ wout, float* __restrict__ partial) {
  extern __shared__ float lds[];
  float* W1T  = lds;                   // [64][128]  W1 transposed (K-major)
  float* W2T  = W1T + 64 * 128;        // [128][64]
  float* ACTB = W2T + 128 * 64;        // 4 waves * 16*65 (padded)
  float* H1B  = ACTB + 4 * 16 * 65;    // 4 waves * 16*129 (padded)
  float* STS  = H1B + 4 * 16 * 129;    // 2048: [128 cols][8 slots] sums + sqs

  const int tid = threadIdx.x, wave = tid >> 5, lane = tid & 31;
  const int half = lane >> 4, l16 = lane & 15;

  for (int i = tid; i < 128 * 64; i += blockDim.x) { int j = i >> 6, k = i & 63; W1T[k * 128 + j] = w1[i]; }
  if (MODE >= 1)
    for (int i = tid; i < 64 * 128; i += blockDim.x) { int j = i >> 7, k = i & 127; W2T[k * 64 + j] = w2[i]; }
  __syncthreads();

  float* actW = ACTB + wave * (16 * 65);
  float* h1W  = H1B + wave * (16 * 129);

  float accS[8], accQ[8];
#pragma unroll
  for (int i = 0; i < 8; i++) { accS[i] = 0.f; accQ[i] = 0.f; }

  const int nwaves = gridDim.x * (blockDim.x >> 5);
  for (int tile = blockIdx.x * (blockDim.x >> 5) + wave; tile < GN / 16; tile += nwaves) {
    const int rowbase = tile * 16;
    // stage 16x64 activation tile: lane -> row l16, cols [half*32, half*32+32)
    {
      int row = rowbase + l16;
      int g = row / NG, n = row - g * NG;
      float f = x[(size_t)2 * row + 1];
      const float* bp = baseF + n * HID;
      const float* ep = embT + g * HID;
#pragma unroll
      for (int k = 0; k < 32; k++) {
        int h = half * 32 + k;
        float c = bp[h] + ep[h] + pww[h] * f + pwb[h];
        float y = fmaxf(c * s1[h] + t1[h], 0.f);
        actW[l16 * 65 + h] = fmaxf(y * s2[h] + t2[h], 0.f);
      }
    }
    asm volatile("s_wait_dscnt 0" ::: "memory");

    // layer1: h1pre = act @ W1^T, 8 column tiles of 16, K=64 in steps of 4
#pragma unroll 1
    for (int tc = 0; tc < 8; tc++) {
      v8f c = {};
#pragma unroll
      for (int k0 = 0; k0 < 64; k0 += 4) {
        int ka = k0 + half * 2;
        v2f a, bf;
        a.x = actW[l16 * 65 + ka];
        a.y = actW[l16 * 65 + ka + 1];
        bf.x = W1T[ka * 128 + tc * 16 + l16];
        bf.y = W1T[(ka + 1) * 128 + tc * 16 + l16];
        c = __builtin_amdgcn_wmma_f32_16x16x4_f32(false, a, false, bf, (short)0, c, false, false);
      }
      int col = tc * 16 + l16;
      float bias = b1[col];
      if (MODE == 0) {
#pragma unroll
        for (int i = 0; i < 8; i++) { float v = c[i] + bias; accS[tc] += v; accQ[tc] += v * v; }
      } else {
        float sc = r1s[col], sh = r1t[col];
#pragma unroll
        for (int i = 0; i < 8; i++)
          h1W[(half * 8 + i) * 129 + col] = fmaxf((c[i] + bias) * sc + sh, 0.f);
      }
    }

    if (MODE >= 1) {
      asm volatile("s_wait_dscnt 0" ::: "memory");
      // layer2: out = h1 @ W2^T, 4 column tiles, K=128
#pragma unroll 1
      for (int tc = 0; tc < 4; tc++) {
        v8f c = {};
#pragma unroll
        for (int k0 = 0; k0 < 128; k0 += 4) {
          int ka = k0 + half * 2;
          v2f a, bf;
          a.x = h1W[l16 * 129 + ka];
          a.y = h1W[l16 * 129 + ka + 1];
          bf.x = W2T[ka * 64 + tc * 16 + l16];
          bf.y = W2T[(ka + 1) * 64 + tc * 16 + l16];
          c = __builtin_amdgcn_wmma_f32_16x16x4_f32(false, a, false, bf, (short)0, c, false, false);
        }
        int col = tc * 16 + l16;
        float bias = b2[col];
        if (MODE == 1) {
#pragma unroll
          for (int i = 0; i < 8; i++) { float v = c[i] + bias; accS[tc] += v; accQ[tc] += v * v; }
        } else {
          float sc = r2s[col], sh = r2t[col];
#pragma unroll
          for (int i = 0; i < 8; i++) actW[(half * 8 + i) * 65 + col] = (c[i] + bias) * sc + sh;
        }
      }
      if (MODE == 2) {
        asm volatile("s_wait_dscnt 0" ::: "memory");
        if (lane < 16) {
          int row = rowbase + lane;
          int n = row % NG;
          float acc = ib1[n];
          const float* wv = iw1 + n * HID;
          for (int h = 0; h < HID; h++) acc += actW[lane * 65 + h] * wv[h];
          wout[row] = acc;
        }
      }
    }
  }

  if (MODE <= 1) {
    const int ntc = (MODE == 0) ? 8 : 4;
    for (int tc = 0; tc < ntc; tc++) {
      int col = tc * 16 + l16;
      STS[col * 8 + wave * 2 + half] = accS[tc];
      STS[1024 + col * 8 + wave * 2 + half] = accQ[tc];
    }
    __syncthreads();
    int ncols = (MODE == 0) ? 128 : 64;
    if (tid < ncols) {
      float S = 0.f, Q = 0.f;
      for (int k = 0; k < 8; k++) { S += STS[tid * 8 + k]; Q += STS[1024 + tid * 8 + k]; }
      partial[blockIdx.x * 2 * ncols + tid] = S;
      partial[blockIdx.x * 2 * ncols + ncols + tid] = Q;
    }
  }
}

// cross_gene layer1: [64,5000] x [64,5000]^T -> [64,64]
__global__ void k_lincg(const float* __restrict__ w, const float* __restrict__ cw,
                        const float* __restrict__ cb, float* __restrict__ dst) {
  int i = blockIdx.x * blockDim.x + threadIdx.x;
  if (i >= GG * HID) return;
  int g = i >> 6, j = i & 63;
  const float* wr = w + (size_t)g * NG;
  const float* cr = cw + (size_t)j * NG;
  float acc = cb[j];
  for (int n = 0; n < NG; n++) acc += wr[n] * cr[n];
  dst[i] = acc;
}

__global__ void k_out2(const float* __restrict__ w, const float* __restrict__ cg,
                       const float* __restrict__ iw2, const float* __restrict__ ib2,
                       const float* __restrict__ x, float* __restrict__ out) {
  int i = blockIdx.x * blockDim.x + threadIdx.x;
  if (i >= GN) return;
  int g = i / NG, n = i - g * NG;
  const float* w2r = iw2 + (size_t)n * 65;
  float acc = w[i] * w2r[0];
  const float* cgr = cg + g * HID;
  for (int h = 0; h < HID; h++) acc += cgr[h] * w2r[1 + h];
  out[i] = acc + ib2[n] + x[(size_t)2 * i];
}

// ---------------- host orchestration ----------------

extern "C" void kernel_launch(void* const* d_in, const int* in_sizes, int n_in,
                              void* d_out, int out_size, void* d_ws, size_t ws_size,
                              hipStream_t stream) {
  auto F = [&](int i) { return (const float*)d_in[i]; };
  const float* x      = F(0);
  const int*   ei_co  = (const int*)d_in[1];
  const float* ew_co  = F(2);
  const int*   ei_go  = (const int*)d_in[3];
  const float* ew_go  = F(4);
  const float* gene_emb = F(5);
  const float* pert_emb = F(6);
  const float* emb_pos  = F(7);
  const float* pw_w = F(8),  *pw_b = F(9);
  const float* sgco_w = F(10), *sgco_b = F(11);
  const float* sggo_w = F(12), *sggo_b = F(13);
  const float* mt_l1w = F(14), *mt_l1b = F(15), *mt_bn1g = F(16), *mt_bn1b = F(17);
  const float* mt_l2w = F(18), *mt_l2b = F(19), *mt_bn2g = F(20), *mt_bn2b = F(21);
  const float* pf_l1w = F(22), *pf_l1b = F(23), *pf_bn1g = F(24), *pf_bn1b = F(25);
  const float* pf_l2w = F(26), *pf_l2b = F(27), *pf_bn2g = F(28), *pf_bn2b = F(29);
  const float* rw_l1w = F(30), *rw_l1b = F(31), *rw_bn1g = F(32), *rw_bn1b = F(33);
  const float* rw_l2w = F(34), *rw_l2b = F(35), *rw_bn2g = F(36), *rw_bn2b = F(37);
  const float* cg_l1w = F(38), *cg_l1b = F(39), *cg_bn1g = F(40), *cg_bn1b = F(41);
  const float* cg_l2w = F(42), *cg_l2b = F(43), *cg_bn2g = F(44), *cg_bn2b = F(45);
  const float* iw1 = F(46), *ib1 = F(47), *iw2 = F(48), *ib2 = F(49);
  const float* bnE_g = F(50), *bnE_b = F(51);
  const float* bnP_g = F(52), *bnP_b = F(53);
  const float* bnT_g = F(54), *bnT_b = F(55);

  char* wsp = (char*)d_ws;
  auto alloc = [&](size_t nf) -> float* {
    float* p = (float*)wsp;
    wsp += ((nf * 4 + 255) / 256) * 256;
    return p;
  };
  float* tge  = alloc(NG * HID);   // renorm(gene_emb)
  float* tpe  = alloc(NG * HID);   // renorm(pert_emb)
  float* tep  = alloc(NG * HID);   // renorm(emb_pos)
  float* base0 = alloc(NG * HID);
  float* deg  = alloc(NG);
  float* dinv = alloc(NG);
  float* agg  = alloc(NG * HID);
  float* baseA = alloc(NG * HID);
  float* tmpA = alloc(NG * HID);
  float* tmpB = alloc(NG * HID);
  float* baseF = alloc(NG * HID);
  float* aggG = alloc(NG * HID);
  float* pglob = alloc(NG * HID);
  float* psum = alloc(GG * HID);
  float* pf1  = alloc(GG * HID);
  float* pf2  = alloc(GG * HID);
  float* embT = alloc(GG * HID);
  float* sE = alloc(64), *tE = alloc(64);
  float* sM1 = alloc(64), *tM1 = alloc(64), *sM2 = alloc(64), *tM2 = alloc(64);
  float* sP1 = alloc(64), *tP1 = alloc(64), *sP2 = alloc(64), *tP2 = alloc(64);
  float* s1 = alloc(64), *t1 = alloc(64), *s2 = alloc(64), *t2 = alloc(64);
  float* r1s = alloc(128), *r1t = alloc(128), *r2s = alloc(64), *r2t = alloc(64);
  float* sC1 = alloc(64), *tC1 = alloc(64), *sC2 = alloc(64), *tC2 = alloc(64);
  float* partialBig = alloc(512 * 128);
  float* partialRec = alloc(500 * 256);
  float* warr = alloc(GN);
  float* cg1 = alloc(GG * HID);
  float* cg2 = alloc(GG * HID);
  float* cg3 = alloc(GG * HID);

  const int B = 256;
  auto gr = [&](int n) { return (n + B - 1) / B; };

  // ---- node path: base = relu(bn_emb(renorm(gene_emb))) + 0.2*sgconv_co ----
  k_renorm<<<gr(NG), B, 0, stream>>>(gene_emb, tge, NG);
  k_renorm<<<gr(NG), B, 0, stream>>>(pert_emb, tpe, NG);
  k_renorm<<<gr(NG), B, 0, stream>>>(emb_pos, tep, NG);
  k_bnstats<<<64, 256, 0, stream>>>(tge, NG, bnE_g, bnE_b, sE, tE);
  k_bnapply<<<gr(NG * HID), B, 0, stream>>>(tge, base0, NG * HID, sE, tE, 1);
  k_zero<<<gr(NG), B, 0, stream>>>(deg, NG);
  k_scatter_deg<<<gr(NE), B, 0, stream>>>(ei_co, ew_co, deg, NE);
  k_dinv<<<gr(NG), B, 0, stream>>>(deg, dinv, NG);
  k_selfagg<<<gr(NG * HID), B, 0, stream>>>(tep, dinv, agg, NG);
  k_scatter_edges<<<gr(NE), B, 0, stream>>>(ei_co, ew_co, dinv, tep, agg, NE);
  k_lin64<<<gr(NG * HID), B, 0, stream>>>(agg, sgco_w, sgco_b, base0, 0.2f, baseA, NG);
  // emb_trans_v2 MLP
  k_lin64<<<gr(NG * HID), B, 0, stream>>>(baseA, mt_l1w, mt_l1b, nullptr, 0.f, tmpA, NG);
  k_bnstats<<<64, 256, 0, stream>>>(tmpA, NG, mt_bn1g, mt_bn1b, sM1, tM1);
  k_bnapply<<<gr(NG * HID), B, 0, stream>>>(tmpA, tmpB, NG * HID, sM1, tM1, 1);
  k_lin64<<<gr(NG * HID), B, 0, stream>>>(tmpB, mt_l2w, mt_l2b, nullptr, 0.f, tmpA, NG);
  k_bnstats<<<64, 256, 0, stream>>>(tmpA, NG, mt_bn2g, mt_bn2b, sM2, tM2);
  k_bnapply<<<gr(NG * HID), B, 0, stream>>>(tmpA, baseF, NG * HID, sM2, tM2, 1);

  // ---- perturbation path ----
  k_zero<<<gr(NG), B, 0, stream>>>(deg, NG);
  k_scatter_deg<<<gr(NE), B, 0, stream>>>(ei_go, ew_go, deg, NE);
  k_dinv<<<gr(NG), B, 0, stream>>>(deg, dinv, NG);
  k_selfagg<<<gr(NG * HID), B, 0, stream>>>(tpe, dinv, aggG, NG);
  k_scatter_edges<<<gr(NE), B, 0, stream>>>(ei_go, ew_go, dinv, tpe, aggG, NE);
  k_lin64<<<gr(NG * HID), B, 0, stream>>>(aggG, sggo_w, sggo_b, nullptr, 0.f, pglob, NG);
  k_pertsum<<<GG, HID, 0, stream>>>(x, pglob, psum);
  k_lin64<<<gr(GG * HID), B, 0, stream>>>(psum, pf_l1w, pf_l1b, nullptr, 0.f, pf1, GG);
  k_bnstats<<<64, 256, 0, stream>>>(pf1, GG, pf_bn1g, pf_bn1b, sP1, tP1);
  k_bnapply<<<gr(GG * HID), B, 0, stream>>>(pf1, pf2, GG * HID, sP1, tP1, 1);
  k_lin64<<<gr(GG * HID), B, 0, stream>>>(pf2, pf_l2w, pf_l2b, nullptr, 0.f, pf1, GG);
  k_bnstats<<<64, 256, 0, stream>>>(pf1, GG, pf_bn2g, pf_bn2b, sP2, tP2);
  k_bnapply<<<gr(GG * HID), B, 0, stream>>>(pf1, embT, GG * HID, sP2, tP2, 1);

  // ---- big-batch BN stats (on the fly, nothing materialized) ----
  k_bigstats<<<512, 256, 0, stream>>>(baseF, embT, x, pw_w, pw_b, nullptr, nullptr, 0, partialBig);
  k_bnfinal<<<1, 64, 0, stream>>>(partialBig, 512, 64, (float)GN, bnT_g, bnT_b, s1, t1);
  k_bigstats<<<512, 256, 0, stream>>>(baseF, embT, x, pw_w, pw_b, s1, t1, 1, partialBig);
  k_bnfinal<<<1, 64, 0, stream>>>(partialBig, 512, 64, (float)GN, bnP_g, bnP_b, s2, t2);

  // ---- fused recovery MLP (WMMA) : 3 passes, no [GN,*] materialization ----
  const size_t recLds = (size_t)(64 * 128 + 128 * 64 + 4 * 16 * 65 + 4 * 16 * 129 + 2048) * 4;
  k_recovery<0><<<500, 128, recLds, stream>>>(baseF, embT, x, pw_w, pw_b, s1, t1, s2, t2,
      rw_l1w, rw_l1b, r1s, r1t, rw_l2w, rw_l2b, r2s, r2t, iw1, ib1, warr, partialRec);
  k_bnfinal<<<1, 128, 0, stream>>>(partialRec, 500, 128, (float)GN, rw_bn1g, rw_bn1b, r1s, r1t);
  k_recovery<1><<<500, 128, recLds, stream>>>(baseF, embT, x, pw_w, pw_b, s1, t1, s2, t2,
      rw_l1w, rw_l1b, r1s, r1t, rw_l2w, rw_l2b, r2s, r2t, iw1, ib1, warr, partialRec);
  k_bnfinal<<<1, 64, 0, stream>>>(partialRec, 500, 64, (float)GN, rw_bn2g, rw_bn2b, r2s, r2t);
  k_recovery<2><<<500, 128, recLds, stream>>>(baseF, embT, x, pw_w, pw_b, s1, t1, s2, t2,
      rw_l1w, rw_l1b, r1s, r1t, rw_l2w, rw_l2b, r2s, r2t, iw1, ib1, warr, partialRec);

  // ---- cross_gene_state MLP on w [64,5000] ----
  k_lincg<<<gr(GG * HID), B, 0, stream>>>(warr, cg_l1w, cg_l1b, cg1);
  k_bnstats<<<64, 256, 0, stream>>>(cg1, GG, cg_bn1g, cg_bn1b, sC1, tC1);
  k_bnapply<<<gr(GG * HID), B, 0, stream>>>(cg1, cg2, GG * HID, sC1, tC1, 1);
  k_lin64<<<gr(GG * HID), B, 0, stream>>>(cg2, cg_l2w, cg_l2b, nullptr, 0.f, cg1, GG);
  k_bnstats<<<64, 256, 0, stream>>>(cg1, GG, cg_bn2g, cg_bn2b, sC2, tC2);
  k_bnapply<<<gr(GG * HID), B, 0, stream>>>(cg1, cg3, GG * HID, sC2, tC2, 0);

  // ---- final per-gene combine ----
  k_out2<<<gr(GN), B, 0, stream>>>(warr, cg3, iw2, ib2, x, (float*)d_out);
}